// TSDE_AD_48790828482956
// MI455X (gfx1250) — compile-verified
//
#include <hip/hip_runtime.h>
#include <hip/hip_bf16.h>
#include <math.h>

typedef float v2f __attribute__((ext_vector_type(2)));
typedef float v4f __attribute__((ext_vector_type(4)));
typedef float v8f __attribute__((ext_vector_type(8)));

#define BB   64
#define NN   1024          // patches per batch
#define DD   512           // patch dim (16*32)
#define CL   64            // clusters
#define CSTR (DD + 4)      // padded LDS row stride (floats) -> conflict-free B reads
#define KFAR 51            // 1024 // 20
#define EPSF 1e-12f

// One workgroup = 8 waves = 128 rows of one batch. Each wave: 16 rows x 64 clusters.
__global__ __launch_bounds__(256) void min_dist_kernel(const float* __restrict__ data,
                                                       const float* __restrict__ mask,
                                                       float* __restrict__ out_md) {
    __shared__ __align__(16) float sC[CL * CSTR];   // masked centers (~129 KB of the 320 KB WGP LDS)
    __shared__ float sC2[CL];                       // center squared norms

    const int tid = threadIdx.x;
    const int b   = blockIdx.x >> 3;   // batch
    const int rg  = blockIdx.x & 7;    // 128-row group
    const size_t bbase = (size_t)b * NN * DD;

    // Stage masked centers (first 64 patches of this batch) into LDS: b128 loads/stores.
    for (int i = tid; i < CL * DD / 4; i += 256) {
        const int i4 = i * 4;
        const int c = i4 >> 9, j = i4 & (DD - 1);
        v4f d = *(const v4f*)(data + bbase + i4);
        v4f m = *(const v4f*)(mask + bbase + i4);
        *(v4f*)&sC[c * CSTR + j] = d * m;
    }
    __syncthreads();
    // c2 norms (one thread per center, skewed LDS reads)
    if (tid < CL) {
        float s = 0.f;
        const int skew = tid * 8;
        for (int j = 0; j < DD; ++j) {
            float v = sC[tid * CSTR + ((j + skew) & (DD - 1))];
            s += v * v;
        }
        sC2[tid] = s;
    }
    __syncthreads();

    const int lane = tid & 31;
    const int wave = tid >> 5;
    const int half = lane >> 4;   // 0: K=0,1 | 1: K=2,3 (f32 16x4 A layout)
    const int lrow = lane & 15;   // A row / B col within tile
    const int r0   = rg * 128 + wave * 16;

    const size_t rowoff = bbase + (size_t)(r0 + lrow) * DD + half * 2;
    const float* dp = data + rowoff;
    const float* mp = mask + rowoff;
    const float* c0 = &sC[( 0 + lrow) * CSTR + half * 2];
    const float* c1 = &sC[(16 + lrow) * CSTR + half * 2];
    const float* c2p = &sC[(32 + lrow) * CSTR + half * 2];
    const float* c3 = &sC[(48 + lrow) * CSTR + half * 2];

    v8f acc0 = {}, acc1 = {}, acc2 = {}, acc3 = {};
    float sq = 0.f;   // per-lane partial sum of squares for p2

    for (int kb = 0; kb < DD; kb += 64) {
        // Pull the next 256B of each A stream into the near (WGP) caches while WMMAs run.
        // locality 3 -> WGP-scope prefetch ("pull into all cache levels on miss").
        __builtin_prefetch(dp + kb + 64, 0, 3);
        __builtin_prefetch(mp + kb + 64, 0, 3);
#pragma unroll 4
        for (int kk = 0; kk < 64; kk += 4) {
            const int k = kb + kk;
            v2f xd = *(const v2f*)(dp + k);
            v2f xm = *(const v2f*)(mp + k);
            v2f a;  a.x = xd.x * xm.x;  a.y = xd.y * xm.y;
            sq += a.x * a.x + a.y * a.y;
            v2f b0 = *(const v2f*)(c0 + k);
            v2f b1 = *(const v2f*)(c1 + k);
            v2f b2 = *(const v2f*)(c2p + k);
            v2f b3 = *(const v2f*)(c3 + k);
            acc0 = __builtin_amdgcn_wmma_f32_16x16x4_f32(false, a, false, b0, (short)0, acc0, false, false);
            acc1 = __builtin_amdgcn_wmma_f32_16x16x4_f32(false, a, false, b1, (short)0, acc1, false, false);
            acc2 = __builtin_amdgcn_wmma_f32_16x16x4_f32(false, a, false, b2, (short)0, acc2, false, false);
            acc3 = __builtin_amdgcn_wmma_f32_16x16x4_f32(false, a, false, b3, (short)0, acc3, false, false);
        }
    }

    // p2 for row (lane&15): combine the two K-halves (wave32 shuffle).
    float p2full = sq + __shfl_xor(sq, 16, 32);

    float c2a = sC2[ 0 + lrow], c2b = sC2[16 + lrow];
    float c2c = sC2[32 + lrow], c2d = sC2[48 + lrow];

    float mind[8];
#pragma unroll
    for (int e = 0; e < 8; ++e) {
        // C/D layout: VGPR e holds row M=e (lanes 0-15) or M=e+8 (lanes 16-31), col N=lane%16
        int rowm = e + half * 8;
        float p2v = __shfl(p2full, rowm, 32);   // row sum lives in lane == rowm
        float d0 = sqrtf(fmaxf(p2v + c2a - 2.f * acc0[e], EPSF));
        float d1 = sqrtf(fmaxf(p2v + c2b - 2.f * acc1[e], EPSF));
        float d2 = sqrtf(fmaxf(p2v + c2c - 2.f * acc2[e], EPSF));
        float d3 = sqrtf(fmaxf(p2v + c2d - 2.f * acc3[e], EPSF));
        mind[e] = fminf(fminf(d0, d1), fminf(d2, d3));
    }
    // min over the 16 columns held in this lane-half
#pragma unroll
    for (int e = 0; e < 8; ++e) {
        float m = mind[e];
        m = fminf(m, __shfl_xor(m, 8, 32));
        m = fminf(m, __shfl_xor(m, 4, 32));
        m = fminf(m, __shfl_xor(m, 2, 32));
        m = fminf(m, __shfl_xor(m, 1, 32));
        mind[e] = m;
    }
    if (lrow == 0) {
        int rowbase = b * NN + r0 + half * 8;
#pragma unroll
        for (int e = 0; e < 8; ++e) out_md[rowbase + e] = mind[e];
    }
}

// Deterministic iterative top-51 per batch; JAX tie-break (lower index first).
// The md row (4 KB) is staged global->LDS with the CDNA5 async-to-LDS path:
// 256 lanes x 16 B, tracked on ASYNCcnt.
__global__ __launch_bounds__(256) void topk_kernel(const float* __restrict__ md,
                                                   int* __restrict__ out_idx) {
    __shared__ __align__(16) float vals[NN];
    __shared__ float rv[256];
    __shared__ int   ri[256];
    const int b = blockIdx.x, tid = threadIdx.x;

    {
        // Low 32 bits of a generic LDS address are the LDS byte offset (ISA 10.2).
        unsigned lds_dst = (unsigned)(uintptr_t)(&vals[tid * 4]);
        const float* src = md + (size_t)b * NN + tid * 4;
        asm volatile("global_load_async_to_lds_b128 %0, %1, off"
                     :: "v"(lds_dst), "v"(src) : "memory");
        asm volatile("s_wait_asynccnt 0" ::: "memory");
    }
    __syncthreads();

    for (int kk = 0; kk < KFAR; ++kk) {
        float bv = -INFINITY; int bi = 0x7fffffff;
        for (int i = tid; i < NN; i += 256) {
            float v = vals[i];
            if (v > bv || (v == bv && i < bi)) { bv = v; bi = i; }
        }
        rv[tid] = bv; ri[tid] = bi;
        __syncthreads();
        for (int s = 128; s > 0; s >>= 1) {
            if (tid < s) {
                float ov = rv[tid + s]; int oi = ri[tid + s];
                if (ov > rv[tid] || (ov == rv[tid] && oi < ri[tid])) { rv[tid] = ov; ri[tid] = oi; }
            }
            __syncthreads();
        }
        if (tid == 0) { out_idx[b * KFAR + kk] = ri[0]; vals[ri[0]] = -INFINITY; }
        __syncthreads();
    }
}

extern "C" void kernel_launch(void* const* d_in, const int* in_sizes, int n_in,
                              void* d_out, int out_size, void* d_ws, size_t ws_size,
                              hipStream_t stream) {
    (void)in_sizes; (void)n_in; (void)d_ws; (void)ws_size; (void)out_size;
    const float* data = (const float*)d_in[0];
    const float* mask = (const float*)d_in[1];
    int*   out_idx = (int*)d_out;                    // (64, 51) int32 first
    float* out_md  = (float*)d_out + BB * KFAR;      // (64, 1024) f32 second

    min_dist_kernel<<<dim3(BB * 8), dim3(256), 0, stream>>>(data, mask, out_md);
    topk_kernel<<<dim3(BB), dim3(256), 0, stream>>>(out_md, out_idx);
}